// EdgeModel_1_23630910063280
// MI455X (gfx1250) — compile-verified
//
#include <hip/hip_runtime.h>
#include <hip/hip_bf16.h>

typedef __attribute__((ext_vector_type(2))) float v2f;
typedef __attribute__((ext_vector_type(8))) float v8f;

#define IN_F   9
#define OUT_F  84
#define COL_TILES 6   // 6*16 = 96 >= 84

// ---------------------------------------------------------------------------
// Kernel 1: per-node GEMMs via V_WMMA_F32_16X16X4_F32.
//   p      = x @ W      (W[i][o]   = nn_w[i*84+o])
//   q      = x @ B      (B[i][o]   = nn_b[i*84+o])
//   outacc = x @ root + bias
// One wave (32 threads) per 16-node tile. N = 50000 = 3125 * 16 exactly.
//
// K is padded 9->12. Guards are ARITHMETIC (clamped in-bounds address +
// 0/1 mask multiply), never exec-masked loads, so no branchy load sequences.
// Per K-step k0 = ks*4 + half*2:
//   ks=0: k={0,1|2,3}   all valid
//   ks=1: k={4,5|6,7}   all valid
//   ks=2: k={8,9|10,11} only k=8 (half==0, .x) valid
// ---------------------------------------------------------------------------
template <bool HAS_BIAS>
__device__ __forceinline__ void gemm_mat_tiles(const v2f a[3],
                                               const float* __restrict__ mat,
                                               float* __restrict__ out,
                                               const float* __restrict__ bias,
                                               int nb, int lm, int half) {
    const float mh = (half == 0) ? 1.0f : 0.0f;  // mask for k==8 slot
#pragma unroll
    for (int ct = 0; ct < COL_TILES; ++ct) {
        const int col    = ct * 16 + lm;
        const bool cok   = (col < OUT_F);
        const int  colc  = cok ? col : (OUT_F - 1);     // clamped, in-bounds
        const float cm   = cok ? 1.0f : 0.0f;           // column mask
        v8f acc = {};
#pragma unroll
        for (int ks = 0; ks < 3; ++ks) {
            const int k0 = ks * 4 + half * 2;
            v2f b;
            if (ks < 2) {
                b.x = mat[(k0    ) * OUT_F + colc] * cm;
                b.y = mat[(k0 + 1) * OUT_F + colc] * cm;
            } else {
                // only k=8 row exists; half==1 lanes read row 0 but mask to 0
                b.x = mat[(half ? 0 : 8) * OUT_F + colc] * (cm * mh);
                b.y = 0.0f;
            }
            acc = __builtin_amdgcn_wmma_f32_16x16x4_f32(
                /*neg_a=*/false, a[ks], /*neg_b=*/false, b,
                /*c_mod=*/(short)0, acc, /*reuse_a=*/false, /*reuse_b=*/false);
        }
        if (cok) {
            float bv = 0.0f;
            if (HAS_BIAS) bv = bias[col];
#pragma unroll
            for (int r = 0; r < 8; ++r) {
                const int node = nb + r + half * 8;     // C/D: M = r + 8*(lane/16)
                out[(size_t)node * OUT_F + col] = acc[r] + bv;
            }
        }
    }
}

__global__ __launch_bounds__(32) void node_gemm_kernel(
    const float* __restrict__ x,
    const float* __restrict__ nn_w, const float* __restrict__ nn_b,
    const float* __restrict__ root, const float* __restrict__ bias,
    float* __restrict__ p, float* __restrict__ q, float* __restrict__ outacc) {
    const int lane = threadIdx.x;        // 0..31
    const int lm   = lane & 15;
    const int half = lane >> 4;          // 0 or 1
    const int nb   = blockIdx.x * 16;    // node tile base
    const int node = nb + lm;
    const float mh = (half == 0) ? 1.0f : 0.0f;

    // A tile: 16 nodes x (K padded to 12). Lane holds M=lm; .x=K=k0, .y=K=k0+1.
    v2f a[3];
    {
        const size_t base = (size_t)node * IN_F;
        const int k0a = half * 2;                  // {0,2}
        a[0].x = x[base + k0a    ];
        a[0].y = x[base + k0a + 1];
        a[1].x = x[base + k0a + 4];
        a[1].y = x[base + k0a + 5];
        // ks=2: k = {8,9 | 10,11}; only k=8 valid (half==0 .x)
        a[2].x = x[base + (half ? 0 : 8)] * mh;
        a[2].y = 0.0f;
    }

    gemm_mat_tiles<false>(a, nn_w, p,      nullptr, nb, lm, half);
    gemm_mat_tiles<false>(a, nn_b, q,      nullptr, nb, lm, half);
    gemm_mat_tiles<true >(a, root, outacc, bias,    nb, lm, half);
}

// ---------------------------------------------------------------------------
// Kernel 2: edge gather + scatter-add.
//   outacc[dst,:] += ea[e] * p[src,:] + q[src,:]
// One thread per (edge, 4-column group): 21 groups of 4 cover 84 columns.
// p/q rows are 16B-aligned (84 % 4 == 0), so float4 loads are legal.
// p/q/outacc (16.8 MB each) are L2-resident; atomics resolve in L2.
// ---------------------------------------------------------------------------
__global__ void edge_scatter_kernel(const int* __restrict__ ei,      // [2,E] row-major
                                    const float* __restrict__ ea,    // [E]
                                    const float* __restrict__ p,
                                    const float* __restrict__ q,
                                    float* __restrict__ outacc,
                                    int E_) {
    const int gid = blockIdx.x * blockDim.x + threadIdx.x;
    if (gid >= E_ * 21) return;
    const int e = gid / 21;
    const int j = (gid - e * 21) * 4;          // column base 0,4,...,80
    const int src = ei[e];
    const int dst = ei[E_ + e];
    const float w = ea[e];

    const float4 pv = *reinterpret_cast<const float4*>(p + (size_t)src * OUT_F + j);
    const float4 qv = *reinterpret_cast<const float4*>(q + (size_t)src * OUT_F + j);
    float* o = outacc + (size_t)dst * OUT_F + j;
    atomicAdd(o + 0, fmaf(w, pv.x, qv.x));
    atomicAdd(o + 1, fmaf(w, pv.y, qv.y));
    atomicAdd(o + 2, fmaf(w, pv.z, qv.z));
    atomicAdd(o + 3, fmaf(w, pv.w, qv.w));
}

// ---------------------------------------------------------------------------
// Kernel 3a: column sums / sums of squares in f64.
// One thread per column (84 of 96 lanes active); consecutive lanes read
// consecutive addresses -> coalesced 336B row reads.
// ---------------------------------------------------------------------------
__global__ __launch_bounds__(96) void bn_stats_kernel(const float* __restrict__ outacc,
                                                      double* __restrict__ stats, // [2*84]
                                                      int N_, int rowsPerBlock) {
    const int c = threadIdx.x;
    if (c >= OUT_F) return;
    const int r0 = blockIdx.x * rowsPerBlock;
    int r1 = r0 + rowsPerBlock; if (r1 > N_) r1 = N_;
    double s = 0.0, s2 = 0.0;
    for (int r = r0; r < r1; ++r) {
        const double v = (double)outacc[(size_t)r * OUT_F + c];
        s  += v;
        s2 += v * v;
    }
    atomicAdd(&stats[c], s);
    atomicAdd(&stats[OUT_F + c], s2);
}

// ---------------------------------------------------------------------------
// Kernel 3b: normalize: out = (v - mean) * rsqrt(var + eps) * gamma + beta
// ---------------------------------------------------------------------------
__global__ void bn_norm_kernel(const float* __restrict__ outacc,
                               const double* __restrict__ stats,
                               const float* __restrict__ gamma,
                               const float* __restrict__ beta,
                               float* __restrict__ out,
                               int total, int N_) {
    const int idx = blockIdx.x * blockDim.x + threadIdx.x;
    if (idx >= total) return;
    const int c = idx % OUT_F;
    const double invN = 1.0 / (double)N_;
    const double mean = stats[c] * invN;
    const double var  = stats[OUT_F + c] * invN - mean * mean;
    const float inv   = rsqrtf((float)var + 1e-5f);
    out[idx] = ((outacc[idx] - (float)mean) * inv) * gamma[c] + beta[c];
}

// ---------------------------------------------------------------------------
extern "C" void kernel_launch(void* const* d_in, const int* in_sizes, int n_in,
                              void* d_out, int out_size, void* d_ws, size_t ws_size,
                              hipStream_t stream) {
    const float* x     = (const float*)d_in[0];   // [N,9]
    const int*   ei    = (const int*)  d_in[1];   // [2,E] int32
    const float* ea    = (const float*)d_in[2];   // [E,1]
    const float* nn_w  = (const float*)d_in[3];   // [1,756]
    const float* nn_b  = (const float*)d_in[4];   // [756]
    const float* root  = (const float*)d_in[5];   // [9,84]
    const float* bias  = (const float*)d_in[6];   // [84]
    const float* gamma = (const float*)d_in[7];   // [84]
    const float* beta  = (const float*)d_in[8];   // [84]
    float*       out   = (float*)d_out;           // [N,84]

    const int N_ = in_sizes[0] / IN_F;            // 50000
    const int E_ = in_sizes[2];                   // 250000
    const size_t nf = (size_t)N_ * OUT_F;

    float*  p      = (float*)d_ws;
    float*  q      = p + nf;
    float*  outacc = q + nf;
    double* stats  = (double*)(outacc + nf);

    // Zero the BN accumulators each launch (graph-capture-safe).
    hipMemsetAsync(stats, 0, 2 * OUT_F * sizeof(double), stream);

    // 1) WMMA node GEMMs: p = xW, q = xB, outacc = x@root + bias
    node_gemm_kernel<<<N_ / 16, 32, 0, stream>>>(x, nn_w, nn_b, root, bias, p, q, outacc);

    // 2) Edge gather + atomic scatter-add into outacc
    const int totalE = E_ * 21;
    edge_scatter_kernel<<<(totalE + 255) / 256, 256, 0, stream>>>(ei, ea, p, q, outacc, E_);

    // 3a) BatchNorm stats (f64 accumulation)
    const int rowsPerBlock = 512;
    const int statBlocks = (N_ + rowsPerBlock - 1) / rowsPerBlock;
    bn_stats_kernel<<<statBlocks, 96, 0, stream>>>(outacc, stats, N_, rowsPerBlock);

    // 3b) Normalize into d_out
    const int total = (int)nf;
    bn_norm_kernel<<<(total + 255) / 256, 256, 0, stream>>>(outacc, stats, gamma, beta, out, total, N_);
}